// Scene_7301444403424
// MI455X (gfx1250) — compile-verified
//
#include <hip/hip_runtime.h>

typedef float v2f __attribute__((ext_vector_type(2)));
typedef float v8f __attribute__((ext_vector_type(8)));

#define BIGF 1e30f
#define EPSF 1e-4f

// DPP16 butterfly on a 64-bit key (two 32-bit halves). CTRL immediates:
//   0xB1  quad_perm [1,0,3,2]  -> lane ^ 1
//   0x4E  quad_perm [2,3,0,1]  -> lane ^ 2
//   0x141 row_half_mirror      -> lane ^ 7 within 8 (acts as ^4 after quad min)
//   0x140 row_mirror           -> lane ^ 15 within 16 (acts as ^8 after above)
template <int CTRL>
static __device__ inline unsigned long long dpp_bfly64(unsigned long long k) {
    unsigned lo = (unsigned)k, hi = (unsigned)(k >> 32);
    lo = (unsigned)__builtin_amdgcn_update_dpp(0, (int)lo, CTRL, 0xF, 0xF, true);
    hi = (unsigned)__builtin_amdgcn_update_dpp(0, (int)hi, CTRL, 0xF, 0xF, true);
    return ((unsigned long long)hi << 32) | lo;
}

// One wave = 16 rays. Lane l owns ray r = (l&7) + 8*(l>>4) (each ray duplicated
// in lanes l and l^8 of the same half-wave). Per bounce, per 16-sphere tile:
//   -b  = WMMA( A=(dx,dy,dz,p.d),       B=(cx,cy,cz,-1), C=0 )
//    c  = WMMA( A=(-2px,-2py,-2pz,p.p), B=(cx,cy,cz, 1), C=(c.c - r^2) )
// then disc = b*b - c, raw v_sqrt, near/far select, 64-bit-key argmin via DPP.

__global__ __launch_bounds__(256) void ray_bounce_wmma(
    const float* __restrict__ pos,   const float* __restrict__ dirs,
    const float* __restrict__ inten_in,
    const float* __restrict__ centers, const float* __restrict__ radii,
    const float* __restrict__ refl,  const int* __restrict__ nb_ptr,
    float* __restrict__ out, int N)
{
    __shared__ float scx[64], scy[64], scz[64], sir[64], srf[64], sccr[64];
    const int tid = threadIdx.x;
    if (tid < 64) {
        float cx = centers[tid * 3 + 0];
        float cy = centers[tid * 3 + 1];
        float cz = centers[tid * 3 + 2];
        float rr = radii[tid];
        scx[tid] = cx; scy[tid] = cy; scz[tid] = cz;
        sir[tid] = __builtin_amdgcn_rcpf(rr);   // 1/r, computed once
        srf[tid] = refl[tid];
        sccr[tid] = cx * cx + cy * cy + cz * cz - rr * rr;
    }
    __syncthreads();

    const int lane = tid & 31;
    const int wave = tid >> 5;
    const int hi   = lane >> 4;                 // which 16-lane half
    const bool hiL = (hi != 0);
    const int ray  = (lane & 7) + 8 * hi;       // owned ray (0..15), duplicated
    int gid = blockIdx.x * 128 + wave * 16 + ray;
    if (gid >= N) gid = N - 1;                  // keep EXEC all-ones for WMMA

    float px = pos[gid * 3 + 0], py = pos[gid * 3 + 1], pz = pos[gid * 3 + 2];
    float dx = dirs[gid * 3 + 0], dy = dirs[gid * 3 + 1], dz = dirs[gid * 3 + 2];
    float inten = inten_in[gid];
    {   // normalize direction once (reference does this before the scan)
        float inv = __builtin_amdgcn_rsqf(dx * dx + dy * dy + dz * dz);
        dx *= inv; dy *= inv; dz *= inv;
    }

    // lane that owns ray m = (lane&15) in this wave (for A-operand gather)
    const int srcLane = (lane & 7) + 16 * ((lane >> 3) & 1);
    const int nloc = lane & 15;                 // sphere column within a tile
    const int nbounces = *nb_ptr;

    for (int bounce = 0; bounce < nbounces; ++bounce) {
        float pd = px * dx + py * dy + pz * dz;
        float pp = px * px + py * py + pz * pz;

        // Build 16x4 f32 A operands (lanes 0-15: K0/K1, lanes 16-31: K2/K3)
        float sdx = __shfl(dx, srcLane, 32), sdy = __shfl(dy, srcLane, 32);
        float sdz = __shfl(dz, srcLane, 32), spd = __shfl(pd, srcLane, 32);
        float spx = __shfl(px, srcLane, 32), spy = __shfl(py, srcLane, 32);
        float spz = __shfl(pz, srcLane, 32), spp = __shfl(pp, srcLane, 32);
        v2f Ab, Ac;
        Ab.x = hiL ? sdz : sdx;
        Ab.y = hiL ? spd : sdy;
        Ac.x = hiL ? (-2.0f * spz) : (-2.0f * spx);
        Ac.y = hiL ? spp : (-2.0f * spy);

        unsigned long long kmin[8];
#pragma unroll
        for (int i = 0; i < 8; ++i) kmin[i] = ~0ull;

#pragma unroll
        for (int T = 0; T < 4; ++T) {           // 4 tiles of 16 spheres
            const int n = nloc + T * 16;
            float cx = scx[n], cy = scy[n], cz = scz[n], ccr = sccr[n];
            // 4x16 f32 B operand (VGPR0: K0/K2, VGPR1: K1/K3 across halves)
            v2f Bb, Bc;
            Bb.x = hiL ? cz : cx;
            Bb.y = hiL ? -1.0f : cy;
            Bc.x = Bb.x;
            Bc.y = hiL ? 1.0f : cy;
            v8f Czero = {};
            v8f Cc;
#pragma unroll
            for (int i = 0; i < 8; ++i) Cc[i] = ccr;

            v8f negb = __builtin_amdgcn_wmma_f32_16x16x4_f32(
                false, Ab, false, Bb, (short)0, Czero, false, false);
            v8f cq = __builtin_amdgcn_wmma_f32_16x16x4_f32(
                false, Ac, false, Bc, (short)0, Cc, false, false);

#pragma unroll
            for (int i = 0; i < 8; ++i) {
                float nb = negb[i];             // = -b
                float cv = cq[i];               // quadratic constant
                float disc = nb * nb - cv;
                float sq = __builtin_amdgcn_sqrtf(fmaxf(disc, 0.0f));
                float tn = nb - sq;             // t_near = -b - sq
                float tf = nb + sq;
                float t = (tn > EPSF) ? tn : tf;
                bool ok = (disc > 0.0f) && (t > EPSF);
                t = ok ? t : BIGF;
                unsigned long long key =
                    ((unsigned long long)__float_as_uint(t) << 32) |
                    (unsigned)n;                // tie -> lowest sphere index
                kmin[i] = (key < kmin[i]) ? key : kmin[i];
            }
        }

        // min-reduce across the 16 sphere-lanes inside each half (DPP16, VALU)
#pragma unroll
        for (int i = 0; i < 8; ++i) {
            unsigned long long o;
            o = dpp_bfly64<0xB1>(kmin[i]);  kmin[i] = (o < kmin[i]) ? o : kmin[i];
            o = dpp_bfly64<0x4E>(kmin[i]);  kmin[i] = (o < kmin[i]) ? o : kmin[i];
            o = dpp_bfly64<0x141>(kmin[i]); kmin[i] = (o < kmin[i]) ? o : kmin[i];
            o = dpp_bfly64<0x140>(kmin[i]); kmin[i] = (o < kmin[i]) ? o : kmin[i];
        }

        // pick accumulator row v = lane&7 (that's this lane's owned ray)
        const int v = lane & 7;
        unsigned long long key = kmin[0];
        key = (v == 1) ? kmin[1] : key;
        key = (v == 2) ? kmin[2] : key;
        key = (v == 3) ? kmin[3] : key;
        key = (v == 4) ? kmin[4] : key;
        key = (v == 5) ? kmin[5] : key;
        key = (v == 6) ? kmin[6] : key;
        key = (v == 7) ? kmin[7] : key;

        float tmin = __uint_as_float((unsigned)(key >> 32));
        int idx = (int)((unsigned)key & 63u);
        bool active = (tmin < 0.5f * BIGF) && (inten > 0.0f);
        float ts = active ? tmin : 0.0f;

        float hx = px + ts * dx, hy = py + ts * dy, hz = pz + ts * dz;
        float cwx = scx[idx], cwy = scy[idx], cwz = scz[idx];
        float irw = sir[idx];
        float fw = srf[idx];
        float nx = (hx - cwx) * irw, ny = (hy - cwy) * irw, nz = (hz - cwz) * irw;
        float nd = dx * nx + dy * ny + dz * nz;
        float rdx = dx - 2.0f * nd * nx;
        float rdy = dy - 2.0f * nd * ny;
        float rdz = dz - 2.0f * nd * nz;

        px = active ? hx : px;  py = active ? hy : py;  pz = active ? hz : pz;
        dx = active ? rdx : dx; dy = active ? rdy : dy; dz = active ? rdz : dz;
        inten = active ? inten * fw : inten;
    }

    // each ray is duplicated in two lanes; let one copy write [N,7] output
    if ((lane & 8) == 0) {
        out[gid * 7 + 0] = px;
        out[gid * 7 + 1] = py;
        out[gid * 7 + 2] = pz;
        out[gid * 7 + 3] = dx;
        out[gid * 7 + 4] = dy;
        out[gid * 7 + 5] = dz;
        out[gid * 7 + 6] = inten;
    }
}

extern "C" void kernel_launch(void* const* d_in, const int* in_sizes, int n_in,
                              void* d_out, int out_size, void* d_ws, size_t ws_size,
                              hipStream_t stream) {
    const float* pos      = (const float*)d_in[0];
    const float* dirs     = (const float*)d_in[1];
    const float* inten    = (const float*)d_in[2];
    const float* centers  = (const float*)d_in[3];
    const float* radii    = (const float*)d_in[4];
    const float* refl     = (const float*)d_in[5];
    const int*   nbounces = (const int*)d_in[6];
    float* out = (float*)d_out;

    const int N = in_sizes[2];                  // intensity has one entry/ray
    const int blocks = (N + 127) / 128;         // 128 rays per 256-thread block
    ray_bounce_wmma<<<blocks, 256, 0, stream>>>(
        pos, dirs, inten, centers, radii, refl, nbounces, out, N);
}